// RGCNLayer_3272765080008
// MI455X (gfx1250) — compile-verified
//
#include <hip/hip_runtime.h>

// ---------------------------------------------------------------------------
// RGCN layer for MI455X (gfx1250, wave32):
//   proj[r] = h @ W[r]            -> V_WMMA_F32_16X16X4_F32 (exact f32)
//   out[dst[e]] += proj[rel[e], src[e]] * norm[e]  -> global_atomic_add_f32
//   out = relu(out)
// proj lives in d_ws (R * N * 128 * 4 = 164 MB; fits the 192 MB L2, so the
// random per-edge gather phase is L2-resident).
//
// LDS layout for W[r]: K-pair interleaved + bank-swizzled so each WMMA B
// operand is one aligned ds_load_b64 (no v_mov shuffles, no bank conflicts):
//   dword(k, n) = (k>>1)*256 + ((2*n + 32*((k>>1)&1)) & 255) + (k&1)
// ---------------------------------------------------------------------------

typedef __attribute__((ext_vector_type(2))) float v2f;
typedef __attribute__((ext_vector_type(8))) float v8f;

#define IN_FEAT 128
#define OUT_FEAT 128
#define WAVES_PER_BLOCK 5   // 5 waves * 16 rows = 80 rows/block; 1250 tiles = 250*5

// ---------------- GEMM: proj[r][m][n] = sum_k h[m][k] * W[r][k][n] ----------
__global__ void __launch_bounds__(32 * WAVES_PER_BLOCK)
rgcn_proj_wmma(const float* __restrict__ h,
               const float* __restrict__ w,
               float* __restrict__ proj,
               int n_nodes) {
  __shared__ float wlds[64 * 256];  // 64 KB: 64 K-pairs x 256 interleaved dwords

  const int r    = blockIdx.y;
  const int tid  = threadIdx.x;
  const int lane = tid & 31;
  const int wave = tid >> 5;
  const int m16  = lane & 15;     // M (A) / N (B,C) index within tile
  const int kh   = lane >> 4;     // K-half select (ISA f32 A/B layout)

  // Stage W[r] (128x128 f32) into K-pair-interleaved, swizzled LDS.
  // Each iteration: read rows k=2kp and k=2kp+1 (float4 each), write two
  // interleaved float4s -> fully vectorized ds_store_b128.
  {
    const float4* wsrc = (const float4*)(w + (size_t)r * IN_FEAT * OUT_FEAT);
    for (int idx = tid; idx < 64 * 32; idx += 32 * WAVES_PER_BLOCK) {
      const int kp   = idx >> 5;       // K-pair 0..63
      const int nblk = idx & 31;       // group of 4 columns
      const float4 r0 = wsrc[(2 * kp) * 32 + nblk];      // k=2kp,   n=4nblk..+3
      const float4 r1 = wsrc[(2 * kp + 1) * 32 + nblk];  // k=2kp+1, n=4nblk..+3
      const int base = kp * 256 + ((8 * nblk + ((kp & 1) << 5)) & 255);
      float4* d = (float4*)&wlds[base];
      d[0] = make_float4(r0.x, r1.x, r0.y, r1.y);
      d[1] = make_float4(r0.z, r1.z, r0.w, r1.w);
    }
  }
  __syncthreads();

  const int mtile = blockIdx.x * WAVES_PER_BLOCK + wave;
  int row  = mtile * 16 + m16;
  int lrow = row < n_nodes ? row : (n_nodes - 1);  // clamp: EXEC stays all-1s

  v8f acc[8];
#pragma unroll
  for (int t = 0; t < 8; ++t)
    acc[t] = (v8f){0.f, 0.f, 0.f, 0.f, 0.f, 0.f, 0.f, 0.f};

  const float* arow = h + (size_t)lrow * IN_FEAT;

  for (int k0 = 0; k0 < IN_FEAT; k0 += 4) {
    const int kb = k0 + 2 * kh;          // this lane's K pair: kb, kb+1
    const int kp = kb >> 1;
    const v2f a = *(const v2f*)(arow + kb);       // global_load_b64, aligned pair
    const int rbase = kp * 256;
    const int sw    = (kp & 1) << 5;
#pragma unroll
    for (int t = 0; t < 8; ++t) {        // 8 N-tiles of 16 -> full 128 cols
      const int n = t * 16 + m16;
      const v2f b = *(const v2f*)&wlds[rbase + ((2 * n + sw) & 255)];  // ds_load_b64
      // 8 args: (neg_a, A, neg_b, B, c_mod, C, reuse_a, reuse_b)
      acc[t] = __builtin_amdgcn_wmma_f32_16x16x4_f32(
          false, a, false, b, (short)0, acc[t], false, false);
    }
  }

  // C/D layout: VGPR j, lanes 0-15 -> M=j, lanes 16-31 -> M=j+8; N = lane&15
  float* pbase = proj + (size_t)r * n_nodes * OUT_FEAT;
#pragma unroll
  for (int j = 0; j < 8; ++j) {
    int m = mtile * 16 + j + 8 * kh;
    if (m < n_nodes) {
      float* orow = pbase + (size_t)m * OUT_FEAT;
#pragma unroll
      for (int t = 0; t < 8; ++t)
        orow[t * 16 + m16] = acc[t][j];
    }
  }
}

// ---------------- Edge scatter: one wave32 per edge -------------------------
__global__ void __launch_bounds__(256)
rgcn_scatter(const float* __restrict__ proj,
             const float* __restrict__ norm,
             const int* __restrict__ src,
             const int* __restrict__ dst,
             const int* __restrict__ rel,
             float* __restrict__ out,
             int n_edges, int n_nodes) {
  const int e = blockIdx.x * (blockDim.x >> 5) + (threadIdx.x >> 5);
  if (e >= n_edges) return;
  const int lane = threadIdx.x & 31;

  const float nv = norm[e];
  const float4* p =
      (const float4*)(proj + ((size_t)rel[e] * n_nodes + src[e]) * OUT_FEAT);
  float* o = out + (size_t)dst[e] * OUT_FEAT;

  float4 v = p[lane];  // 32 lanes x 16B = 128 floats (global_load_b128)
  unsafeAtomicAdd(&o[lane * 4 + 0], v.x * nv);
  unsafeAtomicAdd(&o[lane * 4 + 1], v.y * nv);
  unsafeAtomicAdd(&o[lane * 4 + 2], v.z * nv);
  unsafeAtomicAdd(&o[lane * 4 + 3], v.w * nv);
}

// ---------------- Zero-init and ReLU ---------------------------------------
__global__ void zero_f32(float4* __restrict__ p, int n4) {
  int i = blockIdx.x * blockDim.x + threadIdx.x;
  if (i < n4) p[i] = make_float4(0.f, 0.f, 0.f, 0.f);
}

__global__ void relu_f32(float4* __restrict__ p, int n4) {
  int i = blockIdx.x * blockDim.x + threadIdx.x;
  if (i < n4) {
    float4 v = p[i];
    v.x = fmaxf(v.x, 0.f);
    v.y = fmaxf(v.y, 0.f);
    v.z = fmaxf(v.z, 0.f);
    v.w = fmaxf(v.w, 0.f);
    p[i] = v;
  }
}

// ---------------------------------------------------------------------------
extern "C" void kernel_launch(void* const* d_in, const int* in_sizes, int n_in,
                              void* d_out, int out_size, void* d_ws, size_t ws_size,
                              hipStream_t stream) {
  const float* h    = (const float*)d_in[0];
  const float* w    = (const float*)d_in[1];
  const float* norm = (const float*)d_in[2];
  const int*   src  = (const int*)d_in[3];
  const int*   dst  = (const int*)d_in[4];
  const int*   rel  = (const int*)d_in[5];
  float* out  = (float*)d_out;
  float* proj = (float*)d_ws;  // needs R*N*128*4 bytes (164 MB for this shape)

  const int n_nodes = in_sizes[0] / IN_FEAT;
  const int n_rels  = in_sizes[1] / (IN_FEAT * OUT_FEAT);
  const int n_edges = in_sizes[3];

  // 1) out = 0
  const int n4 = out_size / 4;
  zero_f32<<<(n4 + 255) / 256, 256, 0, stream>>>((float4*)out, n4);

  // 2) proj[r] = h @ W[r]  (WMMA f32)
  const int mtiles = (n_nodes + 15) / 16;
  dim3 grid((mtiles + WAVES_PER_BLOCK - 1) / WAVES_PER_BLOCK, n_rels);
  rgcn_proj_wmma<<<grid, 32 * WAVES_PER_BLOCK, 0, stream>>>(h, w, proj, n_nodes);

  // 3) atomic scatter of edge messages
  const int epb = 256 / 32;  // edges per block (one wave per edge)
  rgcn_scatter<<<(n_edges + epb - 1) / epb, 256, 0, stream>>>(
      proj, norm, src, dst, rel, out, n_edges, n_nodes);

  // 4) relu in place
  relu_f32<<<(n4 + 255) / 256, 256, 0, stream>>>((float4*)out, n4);
}